// DendriticLinear_16441134809125
// MI455X (gfx1250) — compile-verified
//
#include <hip/hip_runtime.h>

// Problem constants (from reference)
#define B_DIM   64
#define OUT_DIM 512
#define IN_DIM  512
#define RES     10
#define DT_C    0.001f

typedef float v2f __attribute__((ext_vector_type(2)));
typedef float v8f __attribute__((ext_vector_type(8)));

__device__ __forceinline__ float sigm2(float x) {
    // sigmoid(x) * MAX_CONSTANT (2.0)
    return 2.0f / (1.0f + __expf(-x));
}

// ---------------------------------------------------------------------------
// Phase 1: adjoint recurrence. One wave32 per output row o; each lane owns 16
// consecutive compartments. Produces M[o,j] = dt * W[o,j] * g[o,j] in d_ws.
// ---------------------------------------------------------------------------
__global__ __launch_bounds__(256)
void dendritic_adjoint_kernel(const float* __restrict__ W,
                              const float* __restrict__ TC,
                              const float* __restrict__ SCc,
                              const float* __restrict__ DD,
                              float* __restrict__ Mws) {
    const int wave = threadIdx.x >> 5;
    const int lane = threadIdx.x & 31;
    const int o    = blockIdx.x * 8 + wave;     // 64 blocks * 8 waves = 512 rows
    const int base = o * IN_DIM + lane * 16;

    float tau[16], sc[16], w[16];
    const float4* W4 = (const float4*)(W   + base);
    const float4* T4 = (const float4*)(TC  + base);
    const float4* S4 = (const float4*)(SCc + base);
#pragma unroll
    for (int q = 0; q < 4; ++q) {
        float4 wv = W4[q];
        float4 tv = T4[q];
        float4 sv = S4[q];
        w[4*q+0] = wv.x; w[4*q+1] = wv.y; w[4*q+2] = wv.z; w[4*q+3] = wv.w;
        tau[4*q+0] = sigm2(tv.x); tau[4*q+1] = sigm2(tv.y);
        tau[4*q+2] = sigm2(tv.z); tau[4*q+3] = sigm2(tv.w);
        sc[4*q+0]  = sigm2(sv.x); sc[4*q+1]  = sigm2(sv.y);
        sc[4*q+2]  = sigm2(sv.z); sc[4*q+3]  = sigm2(sv.w);
    }
    const float ddt = sigm2(DD[o]) * DT_C;      // D = dt * dd (diag, per-row)

    float lic[16], ldc[16], g[16];
#pragma unroll
    for (int j = 0; j < 16; ++j) { lic[j] = 0.0f; ldc[j] = 0.0f; g[j] = 0.0f; }

    for (int t = 0; t < RES; ++t) {
        float u[16];
#pragma unroll
        for (int j = 0; j < 16; ++j) u[j] = ddt * ldc[j] + sc[j];

        // Neighbor sums T(u) and T(lic_old): cross-lane boundary exchange.
        float u_up = __shfl_up(u[15], 1, 32);
        float u_dn = __shfl_down(u[0], 1, 32);
        float l_up = __shfl_up(lic[15], 1, 32);
        float l_dn = __shfl_down(lic[0], 1, 32);
        if (lane == 0)  { u_up = 0.0f; l_up = 0.0f; }
        if (lane == 31) { u_dn = 0.0f; l_dn = 0.0f; }

        float tu[16], tl[16];
        tu[0] = u_up + u[1];
        tl[0] = l_up + lic[1];
#pragma unroll
        for (int j = 1; j < 15; ++j) {
            tu[j] = u[j-1] + u[j+1];
            tl[j] = lic[j-1] + lic[j+1];
        }
        tu[15] = u[14] + u_dn;
        tl[15] = lic[14] + l_dn;

#pragma unroll
        for (int j = 0; j < 16; ++j) {
            const int gj = lane * 16 + j;
            const float coef = (gj == 0 || gj == IN_DIM - 1) ? 1.0f : 2.0f;
            const float nlic = lic[j] + u[j];
            ldc[j] = (tau[j] - coef * sc[j]) * u[j] + sc[j] * (tu[j] + tl[j]);
            lic[j] = nlic;
            g[j]  += nlic;
        }
    }

    float4* M4 = (float4*)(Mws + base);
#pragma unroll
    for (int q = 0; q < 4; ++q) {
        float4 mv;
        mv.x = DT_C * g[4*q+0] * w[4*q+0];
        mv.y = DT_C * g[4*q+1] * w[4*q+1];
        mv.z = DT_C * g[4*q+2] * w[4*q+2];
        mv.w = DT_C * g[4*q+3] * w[4*q+3];
        M4[q] = mv;
    }
}

// ---------------------------------------------------------------------------
// Phase 2: soma = x @ M^T via V_WMMA_F32_16X16X4_F32.
// One wave32 per 16x16 output tile; 128 K-steps of K=4 each.
// ---------------------------------------------------------------------------
__global__ __launch_bounds__(256)
void dendritic_gemm_kernel(const float* __restrict__ X,
                           const float* __restrict__ Mws,
                           float* __restrict__ Out) {
    const int wave = threadIdx.x >> 5;
    const int lane = threadIdx.x & 31;
    const int tile = blockIdx.x * 8 + wave;     // 16 blocks * 8 waves = 128 tiles
    const int bt   = tile >> 5;                 // 0..3   (batch tile)
    const int ot   = tile & 31;                 // 0..31  (out tile)
    const int b0   = bt * 16;
    const int o0   = ot * 16;
    const int half = lane >> 4;                 // 0/1
    const int mn   = lane & 15;

    // A: x[b0+m, k], lane holds K = 2*half, 2*half+1 (float2 contiguous)
    const float* xrow = X   + (b0 + mn) * IN_DIM + 2 * half;
    // B: B[k][n] = M[o0+n, k], same K placement per lane
    const float* mrow = Mws + (o0 + mn) * IN_DIM + 2 * half;

    v8f c = {0.f, 0.f, 0.f, 0.f, 0.f, 0.f, 0.f, 0.f};
#pragma unroll 8
    for (int k0 = 0; k0 < IN_DIM; k0 += 4) {
        v2f a = *(const v2f*)(xrow + k0);
        v2f b = *(const v2f*)(mrow + k0);
        // 8 args: (neg_a, A, neg_b, B, c_mod, C, reuse_a, reuse_b)
        c = __builtin_amdgcn_wmma_f32_16x16x4_f32(
                false, a, false, b, (short)0, c, false, false);
    }

#pragma unroll
    for (int r = 0; r < 8; ++r) {
        const int m = r + 8 * half;
        Out[(b0 + m) * OUT_DIM + o0 + mn] = c[r];
    }
}

// ---------------------------------------------------------------------------
extern "C" void kernel_launch(void* const* d_in, const int* in_sizes, int n_in,
                              void* d_out, int out_size, void* d_ws, size_t ws_size,
                              hipStream_t stream) {
    const float* x   = (const float*)d_in[0];   // (B, IN)
    const float* W   = (const float*)d_in[1];   // (OUT, IN)
    const float* tc  = (const float*)d_in[2];   // (OUT, IN)
    const float* scc = (const float*)d_in[3];   // (OUT, IN)
    const float* dd  = (const float*)d_in[4];   // (OUT, 1)
    float* out = (float*)d_out;                 // (B, OUT)
    float* Mws = (float*)d_ws;                  // OUT*IN floats = 1 MB scratch

    (void)in_sizes; (void)n_in; (void)out_size; (void)ws_size;

    dendritic_adjoint_kernel<<<OUT_DIM / 8, 256, 0, stream>>>(W, tc, scc, dd, Mws);
    dendritic_gemm_kernel<<<(B_DIM / 16) * (OUT_DIM / 16) / 8, 256, 0, stream>>>(x, Mws, out);
}